// TransformLoss_17549236371667
// MI455X (gfx1250) — compile-verified
//
#include <hip/hip_runtime.h>
#include <hip/hip_bf16.h>

// ---------------------------------------------------------------------------
// TransformLoss on MI455X (gfx1250, wave32).
// HBM-bound: 4 x 16384 x 768 x fp32 = 192 MB streamed once -> ~8.3 us floor
// at 23.3 TB/s; ~0.3 GFLOP of VALU work => no WMMA (no matmul; WMMA recast
// would issue more instructions than the fma chain while memory is the wall).
// CDNA5-specific optimization is the data-movement path:
//   global_load_async_to_lds_b128 (ASYNCcnt-tracked, NT hint) double-buffered
//   per wave: compute row r from LDS while row r+stride is in flight.
// Deterministic 2-stage reduction (no float atomics).
// NOTE: tl_main placed first in the TU so the disasm snippet shows it.
// ---------------------------------------------------------------------------

#define DDIM   768
#define BROWS  16384
#define NVEC   6          // 6 float4-chunks per lane * 32 lanes * 4 = 768
#define NELEM  24         // elements per lane
#define WPB    4          // waves per block
#define BLOCK  128
#define MAXNB  512

typedef float f32x4 __attribute__((ext_vector_type(4)));

__device__ __forceinline__ float wave_sum(float x) {
#pragma unroll
  for (int off = 16; off > 0; off >>= 1) x += __shfl_xor(x, off, 32);
  return x;
}

__device__ __forceinline__ float nan_to_num_ref(float x) {
  if (__builtin_isnan(x)) return 0.0f;
  if (__builtin_isinf(x)) return x > 0.0f ? 0.1f : -0.1f;
  return x;
}

// One async b128 copy: per-lane 16B, same immediate offset applies to the
// global and the LDS address (CDNA5 ISA).
#define ASYNC1(ldsb, gptr, OFF)                                                  \
  asm volatile("global_load_async_to_lds_b128 %0, %1, off offset:" OFF           \
               " th:TH_LOAD_NT"                                                  \
               :: "v"(ldsb), "v"(gptr) : "memory")
// 6 copies = one 3 KB row-stream for the whole wave (32 lanes x 16B x 6).
#define ASYNC6(ldsb, gptr)   \
  ASYNC1(ldsb, gptr, "0");   \
  ASYNC1(ldsb, gptr, "512"); \
  ASYNC1(ldsb, gptr, "1024");\
  ASYNC1(ldsb, gptr, "1536");\
  ASYNC1(ldsb, gptr, "2048");\
  ASYNC1(ldsb, gptr, "2560")

__global__ __launch_bounds__(BLOCK)
void tl_main(const float* __restrict__ a,   // init_img_vec
             const float* __restrict__ b,   // next_img_vec
             const float* __restrict__ c,   // init_unclip
             const float* __restrict__ d,   // pred_unclip
             float* __restrict__ part_u,    // [NB][768]
             float* __restrict__ part_m,    // [NB][768]
             int*   __restrict__ cnt,
             int nblocks)
{
  // Single static LDS allocation -> placed at LDS byte offset 0; the raw
  // byte offsets handed to the async asm use the same base as the compiler's
  // ds_load addressing of smem.
  __shared__ float smem[WPB * 2 * 4 * DDIM];     // 96 KB: [wave][parity][stream][768]
  const int tid  = threadIdx.x;
  const int wave = tid >> 5;
  const int lane = tid & 31;
  const int gw   = blockIdx.x * WPB + wave;      // global wave id
  const int nw   = nblocks * WPB;                // total waves
  const int laneOff = lane << 2;                 // float offset of this lane in a row

  float accU[NELEM], accM[NELEM];
#pragma unroll
  for (int e = 0; e < NELEM; ++e) { accU[e] = 0.0f; accM[e] = 0.0f; }
  int cntLocal = 0;

  auto bufIdx = [&](int parity, int stream) -> int {
    return ((wave * 2 + parity) * 4 + stream) * DDIM;   // float index into smem
  };
  auto issueRow = [&](int r, int p) {
    const int ro = r * DDIM + laneOff;
    const unsigned la = (unsigned)((bufIdx(p, 0) + laneOff) * 4);  // LDS byte addr
    const unsigned lb = (unsigned)((bufIdx(p, 1) + laneOff) * 4);
    const unsigned lc = (unsigned)((bufIdx(p, 2) + laneOff) * 4);
    const unsigned ld = (unsigned)((bufIdx(p, 3) + laneOff) * 4);
    ASYNC6(la, a + ro);
    ASYNC6(lb, b + ro);
    ASYNC6(lc, c + ro);
    ASYNC6(ld, d + ro);
  };

  int p = 0;
  if (gw < BROWS) issueRow(gw, 0);               // prologue: first row in flight
  for (int r = gw; r < BROWS; r += nw, p ^= 1) {
    const int rn = r + nw;
    if (rn < BROWS) {
      issueRow(rn, p ^ 1);                       // next row starts flying
      // In-order async completion: <=24 outstanding == the 24 just issued,
      // i.e. the current row's 24 copies have landed.
      asm volatile("s_wait_asynccnt 0x18" ::: "memory");
    } else {
      asm volatile("s_wait_asynccnt 0x0" ::: "memory");
    }

    const float* pa = smem + bufIdx(p, 0);
    const float* pb = smem + bufIdx(p, 1);
    const float* pc = smem + bufIdx(p, 2);
    const float* pd = smem + bufIdx(p, 3);

    float df[NELEM], du[NELEM];
    float qd = 0.f, s12 = 0.f, q22 = 0.f, sd = 0.f, s2 = 0.f;
#pragma unroll
    for (int j = 0; j < NVEC; ++j) {
      const int o = laneOff + j * 128;
      f32x4 va = *(const f32x4*)(pa + o);        // ds_load_b128
      f32x4 vb = *(const f32x4*)(pb + o);
      f32x4 vc = *(const f32x4*)(pc + o);
      f32x4 vd = *(const f32x4*)(pd + o);
      f32x4 f = va - vb;
      f32x4 u = vc - vd;
#pragma unroll
      for (int k = 0; k < 4; ++k) {
        df[4 * j + k] = f[k];
        du[4 * j + k] = u[k];
      }
    }
#pragma unroll
    for (int e = 0; e < NELEM; ++e) {
      const float f = df[e], u = du[e];
      qd  = fmaf(f, f, qd);
      s12 = fmaf(f, u, s12);
      q22 = fmaf(u, u, q22);
      sd += f;
      s2 += u;
      const float dd = f - u;
      accM[e] = fmaf(dd, dd, accM[e]);           // full-batch MSE term
    }
    qd  = wave_sum(qd);
    s12 = wave_sum(s12);
    q22 = wave_sum(q22);
    sd  = wave_sum(sd);
    s2  = wave_sum(s2);

    // mask = ||diff_img|| > 1e-8; masked-out rows contribute 0 to loss_u,
    // so the where(...,1.0) safe branch is dead weight -> skip (wave-uniform).
    const bool mask = qd > 1e-16f;
    if (mask) {
      ++cntLocal;
      const float norm1 = __builtin_sqrtf(qd);
      const float norm2 = __builtin_sqrtf(q22);
      float cosang = s12 / (norm1 * norm2);
      cosang = fminf(1.0f, fmaxf(-1.0f, cosang));
      const float sin_a = __builtin_sqrtf(fmaxf(0.0f, 1.0f - cosang * cosang));
      const float cam1  = cosang - 1.0f;
      const float inv1  = 1.0f / norm1;
      const float d12n  = s12 * inv1;                         // n1 . v2
      const float wn    = __builtin_sqrtf(fmaxf(0.0f, q22 - d12n * d12n));
      const float invw  = 1.0f / wn;                          // degenerate -> NaN -> nan_to_num
      const float s1    = sd * inv1;                          // sum(n1)
      const float s2n   = (s2 - d12n * s1) * invw;            // sum(n2)
#pragma unroll
      for (int e = 0; e < NELEM; ++e) {
        const float n1e = df[e] * inv1;
        const float n2e = (du[e] - n1e * d12n) * invw;
        const float ut  = sin_a * (n2e * s1 - n1e * s2n)
                        + cam1  * (n1e * s1 + n2e * s2n);
        accU[e] = fmaf(ut, ut, accU[e]);
      }
    }
  }
  asm volatile("s_wait_asynccnt 0x0" ::: "memory");  // nothing left in flight

  // Dump accumulators into this wave's own (idle) stream buffers 0/1,
  // then fixed-order cross-wave combine -> deterministic.
  float* su = smem + bufIdx(0, 0);
  float* sm = smem + bufIdx(0, 1);
#pragma unroll
  for (int e = 0; e < NELEM; ++e) {
    const int col = ((e >> 2) << 7) + laneOff + (e & 3);
    su[col] = accU[e];
    sm[col] = accM[e];
  }
  __syncthreads();
  for (int col = tid; col < DDIM; col += BLOCK) {
    float tu = 0.0f, tm = 0.0f;
#pragma unroll
    for (int w = 0; w < WPB; ++w) {
      tu += smem[(w * 8 + 0) * DDIM + col];
      tm += smem[(w * 8 + 1) * DDIM + col];
    }
    part_u[blockIdx.x * DDIM + col] = tu;
    part_m[blockIdx.x * DDIM + col] = tm;
  }
  if (lane == 0 && cntLocal) atomicAdd(cnt, cntLocal);   // integer: deterministic
}

__global__ __launch_bounds__(256)
void tl_reduce(const float* __restrict__ part_u,
               const float* __restrict__ part_m,
               const int*   __restrict__ cnt,
               float* __restrict__ out,
               int nblocks)
{
  const int t = blockIdx.x * 256 + threadIdx.x;
  if (t >= 2 * DDIM) return;
  if (t < DDIM) {
    float s = 0.0f;
    for (int bb = 0; bb < nblocks; ++bb) s += part_u[bb * DDIM + t];
    const float cf = fmaxf((float)(*cnt), 1.0f);
    out[t] = 0.5f * nan_to_num_ref(s / cf);            // WEIGHT_ROTE * nan_to_num(loss_u)
  } else {
    const int col = t - DDIM;
    float s = 0.0f;
    for (int bb = 0; bb < nblocks; ++bb) s += part_m[bb * DDIM + col];
    out[t] = 0.5f * (s / (float)BROWS);                // WEIGHT_MSE * (mean + loss_I(=0))
  }
}

__global__ void tl_zero_cnt(int* cnt) { *cnt = 0; }

extern "C" void kernel_launch(void* const* d_in, const int* in_sizes, int n_in,
                              void* d_out, int out_size, void* d_ws, size_t ws_size,
                              hipStream_t stream) {
  (void)in_sizes; (void)n_in; (void)out_size;
  const float* a = (const float*)d_in[0];
  const float* b = (const float*)d_in[1];
  const float* c = (const float*)d_in[2];
  const float* d = (const float*)d_in[3];

  // Workspace layout: [cnt(int), pad to 64B][part_u NB*768][part_m NB*768]
  int NB = MAXNB;
  const size_t hdr = 64;
  size_t need = hdr + (size_t)NB * DDIM * 2 * sizeof(float);
  if (ws_size < need) {
    NB = (int)((ws_size - hdr) / ((size_t)DDIM * 2 * sizeof(float)));
    if (NB < 1) NB = 1;
  }
  int*   cnt    = (int*)d_ws;
  float* part_u = (float*)((char*)d_ws + hdr);
  float* part_m = part_u + (size_t)NB * DDIM;

  tl_zero_cnt<<<1, 1, 0, stream>>>(cnt);
  tl_main<<<NB, BLOCK, 0, stream>>>(a, b, c, d, part_u, part_m, cnt, NB);
  tl_reduce<<<(2 * DDIM + 255) / 256, 256, 0, stream>>>(part_u, part_m, cnt,
                                                        (float*)d_out, NB);
}